// TokenClassifierWithCRF_31653908971521
// MI455X (gfx1250) — compile-verified
//
#include <hip/hip_runtime.h>

typedef float v2f __attribute__((ext_vector_type(2)));
typedef float v8f __attribute__((ext_vector_type(8)));

#define BATCH 64
#define TLEN  512
#define HDIM  1024
#define NCLS  21

// ---------------------------------------------------------------------------
// Kernel 1: emissions = HS @ W + b  via V_WMMA_F32_16X16X4_F32
// M = 32768 (=B*T), K = 1024, N = 21 (padded to 32).
// 256 blocks x 256 threads (8 waves). Each wave: one 16(M) x 32(N) strip.
// W staged in LDS pre-swizzled into fragment order (one ds_load_b64 per B
// fragment, 64-bank conflict-free). A staged via explicit tmp[8] so all 8
// global_load_b128 are in flight before the ds_store drain; next K-chunk is
// prefetched (global_prefetch_b8) during staging.
// ---------------------------------------------------------------------------
__global__ void __launch_bounds__(256)
gemm_emissions_kernel(const float* __restrict__ HS, const float* __restrict__ W,
                      const float* __restrict__ bias, float* __restrict__ out) {
    // Fragment-ordered W chunk: [ks(16)][nt(2)][half(2)][l16(16)][pair(2)]
    __shared__ float Wlds[16 * 128];       // 8 KB
    __shared__ float Alds[8 * 16 * 68];    // per-wave 16x64 A tile, stride 68 (bank-safe)

    const int tid  = threadIdx.x;
    const int lane = tid & 31;
    const int wave = tid >> 5;
    const int half = lane >> 4;            // 0: lanes 0-15, 1: lanes 16-31
    const int l16  = lane & 15;
    const long mbase = ((long)blockIdx.x * 8 + wave) * 16;
    float* aw = Alds + wave * (16 * 68);

    v8f acc0 = {};   // N columns 0..15
    v8f acc1 = {};   // N columns 16..31 (only 16..20 stored)

    for (int kb = 0; kb < HDIM; kb += 64) {
        // --- W chunk: load phase (8 independent loads), then store phase ---
        float wv[8];
        int   widx[8];
        #pragma unroll
        for (int i = 0; i < 8; ++i) {
            const int idx = i * 256 + tid;         // 0..2047
            const int ks  = idx >> 7;              // k-step 0..15
            const int rem = idx & 127;
            const int nt  = rem >> 6;              // N tile
            const int hf  = (rem >> 5) & 1;        // lane half
            const int l   = (rem >> 1) & 15;       // lane-in-half
            const int p   = rem & 1;               // pair element
            const int krow = ks * 4 + hf * 2 + p;
            const int col  = nt * 16 + l;
            widx[i] = idx;
            wv[i] = (col < NCLS) ? W[(long)(kb + krow) * NCLS + col] : 0.0f;
        }
        #pragma unroll
        for (int i = 0; i < 8; ++i) Wlds[widx[i]] = wv[i];

        // --- A tile: issue all 8 global_load_b128, prefetch next chunk, drain ---
        float4 tmp[8];
        #pragma unroll
        for (int i = 0; i < 8; ++i) {
            const int row = i * 2 + half;          // lanes 0-15 even rows, 16-31 odd
            tmp[i] = *(const float4*)(HS + (mbase + row) * (long)HDIM + kb + l16 * 4);
        }
        if (kb + 64 < HDIM) {
            #pragma unroll
            for (int i = 0; i < 8; ++i) {
                const int row = i * 2 + half;
                __builtin_prefetch(HS + (mbase + row) * (long)HDIM + kb + 64 + l16 * 4, 0, 1);
            }
        }
        #pragma unroll
        for (int i = 0; i < 8; ++i) {
            const int row = i * 2 + half;
            *(float4*)(aw + row * 68 + l16 * 4) = tmp[i];
        }
        __syncthreads();

        // --- 16 k-steps of 4, two N-tiles each ---
        #pragma unroll
        for (int ks = 0; ks < 16; ++ks) {
            // A fragment: lane half selects K pair {kk,kk+1} vs {kk+2,kk+3}
            v2f a = *(const v2f*)(aw + l16 * 68 + ks * 4 + half * 2);
            const float* bp = Wlds + ks * 128 + half * 32 + l16 * 2;
            v2f b0 = *(const v2f*)(bp);            // nt = 0
            v2f b1 = *(const v2f*)(bp + 64);       // nt = 1 (256B apart -> 2addr fuse)
            acc0 = __builtin_amdgcn_wmma_f32_16x16x4_f32(false, a, false, b0,
                                                         (short)0, acc0, false, false);
            acc1 = __builtin_amdgcn_wmma_f32_16x16x4_f32(false, a, false, b1,
                                                         (short)0, acc1, false, false);
        }
        __syncthreads();
    }

    // --- store D + bias; D layout: VGPR r -> M = r + 8*half, N = l16 (+16) ---
    const int c0 = l16;
    const int c1 = 16 + l16;
    const float bc0 = bias[c0];
    const float bc1 = (c1 < NCLS) ? bias[c1] : 0.0f;
    #pragma unroll
    for (int r = 0; r < 8; ++r) {
        const long m = mbase + r + 8 * half;
        out[m * NCLS + c0] = acc0[r] + bc0;
        if (c1 < NCLS) out[m * NCLS + c1] = acc1[r] + bc1;
    }
}

// ---------------------------------------------------------------------------
// Kernel 2: CRF log-likelihood per batch element. One wave32 per batch.
// Lane j owns class j (j<21). alpha broadcast via shuffles.
// ---------------------------------------------------------------------------
__global__ void __launch_bounds__(32)
crf_nll_kernel(const float* __restrict__ em_all, const int* __restrict__ labels,
               const int* __restrict__ amask, const float* __restrict__ trans,
               const float* __restrict__ startT, const float* __restrict__ endT,
               float* __restrict__ partial) {
    const int b = blockIdx.x;
    const int j = threadIdx.x;
    const float NEG = -1e30f;
    const float* em  = em_all + (long)b * TLEN * NCLS;
    const int*   lab = labels + (long)b * TLEN;
    const int*   msk = amask  + (long)b * TLEN;

    // transition column j held in registers
    float tcol[NCLS];
    #pragma unroll
    for (int i = 0; i < NCLS; ++i)
        tcol[i] = (j < NCLS) ? trans[i * NCLS + j] : 0.0f;

    float alpha = (j < NCLS) ? (startT[j] + em[j]) : NEG;

    // ---- forward scan (sequential in t) ----
    for (int t = 1; t < TLEN; ++t) {
        const int   mt   = (msk[t] != 0) && (lab[t] != -100);
        const float em_t = (j < NCLS) ? em[t * NCLS + j] : 0.0f;
        float vals[NCLS];
        #pragma unroll
        for (int i = 0; i < NCLS; ++i)
            vals[i] = __shfl(alpha, i, 32) + tcol[i];
        float m = vals[0];
        #pragma unroll
        for (int i = 1; i < NCLS; ++i) m = fmaxf(m, vals[i]);
        float s = 0.0f;
        #pragma unroll
        for (int i = 0; i < NCLS; ++i) s += __expf(vals[i] - m);
        const float nxt = m + __logf(s) + em_t;
        if (mt && j < NCLS) alpha = nxt;
    }

    // ---- numerator: lanes stride over t ----
    float numer = 0.0f;
    int   cnt   = 0;
    for (int t = 1 + j; t < TLEN; t += 32) {
        if ((msk[t] != 0) && (lab[t] != -100)) {
            const int pv = lab[t - 1], cu = lab[t];
            numer += trans[pv * NCLS + cu] + em[t * NCLS + cu];
            cnt += 1;
        }
    }
    #pragma unroll
    for (int o = 16; o > 0; o >>= 1) {
        numer += __shfl_xor(numer, o, 32);
        cnt   += __shfl_xor(cnt, o, 32);
    }

    // ---- logZ = logsumexp(alphaT + endT) ----
    const float v = (j < NCLS) ? (alpha + endT[j]) : NEG;
    float mm = v;
    #pragma unroll
    for (int o = 16; o > 0; o >>= 1) mm = fmaxf(mm, __shfl_xor(mm, o, 32));
    float e = (j < NCLS) ? __expf(v - mm) : 0.0f;
    #pragma unroll
    for (int o = 16; o > 0; o >>= 1) e += __shfl_xor(e, o, 32);
    const float logZ = mm + __logf(e);

    if (j == 0) {
        const int tag0 = lab[0];
        float num = startT[tag0] + em[tag0] + numer;
        const int last_idx = cnt;          // total mask count = 1 + cnt -> idx = cnt
        const int last_tag = lab[last_idx];
        num += endT[last_tag];
        partial[b] = num - logZ;
    }
}

// ---------------------------------------------------------------------------
// Kernel 3: loss = -mean(partial)
// ---------------------------------------------------------------------------
__global__ void __launch_bounds__(32)
reduce_loss_kernel(const float* __restrict__ partial, float* __restrict__ out_loss) {
    const int j = threadIdx.x;
    float s = partial[j] + partial[j + 32];
    #pragma unroll
    for (int o = 16; o > 0; o >>= 1) s += __shfl_xor(s, o, 32);
    if (j == 0) out_loss[0] = -s / (float)BATCH;
}

// ---------------------------------------------------------------------------
extern "C" void kernel_launch(void* const* d_in, const int* in_sizes, int n_in,
                              void* d_out, int out_size, void* d_ws, size_t ws_size,
                              hipStream_t stream) {
    const float* HS     = (const float*)d_in[0];  // (64,512,1024) f32
    const float* W      = (const float*)d_in[1];  // (1024,21) f32
    const float* bias   = (const float*)d_in[2];  // (21,) f32
    const float* trans  = (const float*)d_in[3];  // (21,21) f32
    const float* startT = (const float*)d_in[4];  // (21,) f32
    const float* endT   = (const float*)d_in[5];  // (21,) f32
    const int*   amask  = (const int*)d_in[6];    // (64,512) bool->int
    const int*   labels = (const int*)d_in[7];    // (64,512) int

    float* out     = (float*)d_out;               // emissions (688128) + loss (1)
    float* partial = (float*)d_ws;                // 64 floats

    gemm_emissions_kernel<<<256, 256, 0, stream>>>(HS, W, bias, out);
    crf_nll_kernel<<<BATCH, 32, 0, stream>>>(out, labels, amask, trans, startT, endT, partial);
    reduce_loss_kernel<<<1, 32, 0, stream>>>(partial, out + (long)BATCH * TLEN * NCLS);
}